// LSM_2688649527627
// MI455X (gfx1250) — compile-verified
//
#include <hip/hip_runtime.h>
#include <hip/hip_bf16.h>

typedef __attribute__((ext_vector_type(2))) float v2f;
typedef __attribute__((ext_vector_type(8))) float v8f;

#define SI_N 6000
#define SJ_N 6000
#define DIM 8
#define EPSF 1e-6f
#define LOG2E 1.44269504088896340736f
#define I_TILES 375          // 6000 / 16
#define J_TILES 375
#define JT_PER_BLOCK 15      // 375 / 25
#define J_BLOCKS 25

// ---------------------------------------------------------------------------
// Zero the scalar output (runs every launch; graph-capture safe).
// ---------------------------------------------------------------------------
__global__ void lsm_init_kernel(float* out) {
    if (threadIdx.x == 0) out[0] = 0.0f;
}

// ---------------------------------------------------------------------------
// Gather sampled rows, pre-scaled by K = log2(e) so the dense epilogue can
// use raw v_exp_f32 (exp2) with no per-element scaling:
//   a_s[t][d] = (latent_zi[si[t]][d] + EPS) * K     na[t] = ||a_s[t]||^2
//   b_s[t][d] =  latent_zj[sj[t]][d] * K            nb[t] = ||b_s[t]||^2
//   beta_s[t] = beta[si[t]] * K,  gamma_s[t] = gamma[sj[t]] * K
// Then K^2*dist^2 = na + nb - 2*(a.b), sqrt -> K*dist, and
// exp(Lambda) = exp2(beta_K + gamma_K - K*dist).
// ---------------------------------------------------------------------------
__global__ void lsm_gather_kernel(const float* __restrict__ beta,
                                  const float* __restrict__ gamma,
                                  const float* __restrict__ zi,
                                  const float* __restrict__ zj,
                                  const int*   __restrict__ si,
                                  const int*   __restrict__ sj,
                                  float* __restrict__ a_s,
                                  float* __restrict__ b_s,
                                  float* __restrict__ beta_s,
                                  float* __restrict__ gamma_s,
                                  float* __restrict__ na,
                                  float* __restrict__ nb) {
    int t = blockIdx.x * blockDim.x + threadIdx.x;
    if (t < SI_N) {
        int idx = si[t];
        float n = 0.0f;
#pragma unroll
        for (int d = 0; d < DIM; ++d) {
            float v = (zi[idx * DIM + d] + EPSF) * LOG2E;
            a_s[t * DIM + d] = v;
            n = fmaf(v, v, n);
        }
        na[t]     = n;
        beta_s[t] = beta[idx] * LOG2E;
    }
    if (t < SJ_N) {
        int idx = sj[t];
        float n = 0.0f;
#pragma unroll
        for (int d = 0; d < DIM; ++d) {
            float v = zj[idx * DIM + d] * LOG2E;
            b_s[t * DIM + d] = v;
            n = fmaf(v, v, n);
        }
        nb[t]      = n;
        gamma_s[t] = gamma[idx] * LOG2E;
    }
}

// ---------------------------------------------------------------------------
// Dense non-link term. One wave (32 lanes) per 16-row i-tile x 15 j-tiles (240 columns).
// Cross term a.b via V_WMMA_F32_16X16X4_F32, K=8 as two chained WMMAs.
//
// f32 16x4 A layout (ISA 7.12.2): lanes 0-15 -> M=lane, V0=K0 V1=K1;
//                                 lanes 16-31 -> M=lane-16, V0=K2 V1=K3.
// f32 4x16 B layout (row striped across lanes): lanes 0-15 -> N=lane, rows
//   K0/K1 in V0/V1; lanes 16-31 -> rows K2/K3.
// C/D 16x16 f32: lane group selects M-half, VGPR r -> M = r + 8*(lane>=16),
//   N = lane & 15.
// ---------------------------------------------------------------------------
__global__ __launch_bounds__(32)
void lsm_dense_kernel(const float* __restrict__ a_s,
                      const float* __restrict__ b_s,
                      const float* __restrict__ beta_s,
                      const float* __restrict__ gamma_s,
                      const float* __restrict__ na,
                      const float* __restrict__ nb,
                      float* __restrict__ out) {
    const int lane = threadIdx.x;     // 0..31, wave32
    const int n    = lane & 15;       // N (and M for A-frag rows)
    const int kh   = lane >> 4;       // K half selector
    const int i0   = blockIdx.x * 16;

    // A fragments for this i-tile: row (i0+n), K split across lane halves.
    const float* arow = a_s + (size_t)(i0 + n) * DIM;
    v2f a0, a1;
    a0.x = arow[kh * 2 + 0];
    a0.y = arow[kh * 2 + 1];
    a1.x = arow[4 + kh * 2 + 0];
    a1.y = arow[4 + kh * 2 + 1];

    // Per-lane row bias / norm for the 8 accumulator rows M = r + 8*kh.
    float row_bias[8], row_norm[8];
#pragma unroll
    for (int r = 0; r < 8; ++r) {
        int m = i0 + r + 8 * kh;
        row_bias[r] = beta_s[m];
        row_norm[r] = na[m];
    }

    float acc = 0.0f;

    const int jt_begin = blockIdx.y * JT_PER_BLOCK;
#pragma unroll 1
    for (int jt = jt_begin; jt < jt_begin + JT_PER_BLOCK; ++jt) {
        const int j0 = jt * 16;
        // B fragment: column (j0+n) of b_s, i.e. B[k][n] = b_s[j0+n][k].
        const float* brow = b_s + (size_t)(j0 + n) * DIM;
        v2f b0, b1;
        b0.x = brow[kh * 2 + 0];
        b0.y = brow[kh * 2 + 1];
        b1.x = brow[4 + kh * 2 + 0];
        b1.y = brow[4 + kh * 2 + 1];

        v8f c = {};
        c = __builtin_amdgcn_wmma_f32_16x16x4_f32(
                false, a0, false, b0, (short)0, c, false, false);
        c = __builtin_amdgcn_wmma_f32_16x16x4_f32(
                false, a1, false, b1, (short)0, c, false, false);

        const float gb  = gamma_s[j0 + n];
        const float nbv = nb[j0 + n];
#pragma unroll
        for (int r = 0; r < 8; ++r) {
            // All quantities pre-scaled by log2(e): d2k = (K*dist)^2.
            float d2k  = row_norm[r] + nbv - 2.0f * c[r];
            d2k        = fmaxf(d2k, 0.0f);
            float lam2 = (row_bias[r] + gb) - __builtin_amdgcn_sqrtf(d2k);
            acc += __builtin_amdgcn_exp2f(lam2);   // == exp(Lambda)
        }
    }

    // Wave32 reduction.
#pragma unroll
    for (int off = 16; off > 0; off >>= 1)
        acc += __shfl_xor(acc, off, 32);
    if (lane == 0)
        atomicAdd(out, -acc);   // LL subtracts sum(exp(Lambda))
}

// ---------------------------------------------------------------------------
// Sparse Poisson link term over NNZ edges.
// ---------------------------------------------------------------------------
__global__ __launch_bounds__(256)
void lsm_links_kernel(const float* __restrict__ beta,
                      const float* __restrict__ gamma,
                      const float* __restrict__ zi,
                      const float* __restrict__ zj,
                      const float* __restrict__ valC,
                      const int*   __restrict__ li,
                      const int*   __restrict__ lj,
                      float* __restrict__ out,
                      int nnz) {
    __shared__ float red[256];
    float acc = 0.0f;
    for (int t = blockIdx.x * blockDim.x + threadIdx.x; t < nnz;
         t += gridDim.x * blockDim.x) {
        const int a = li[t];
        const int b = lj[t];
        float d2 = 0.0f;
#pragma unroll
        for (int d = 0; d < DIM; ++d) {
            float df = zi[a * DIM + d] - zj[b * DIM + d] + EPSF;
            d2 = fmaf(df, df, d2);
        }
        const float v = valC[t];
        acc += v * (beta[a] + gamma[b] - __builtin_amdgcn_sqrtf(d2))
               - lgammaf(v + 1.0f);
    }
    red[threadIdx.x] = acc;
    __syncthreads();
#pragma unroll
    for (int s = 128; s > 0; s >>= 1) {
        if (threadIdx.x < s) red[threadIdx.x] += red[threadIdx.x + s];
        __syncthreads();
    }
    if (threadIdx.x == 0)
        atomicAdd(out, red[0]);
}

// ---------------------------------------------------------------------------
extern "C" void kernel_launch(void* const* d_in, const int* in_sizes, int n_in,
                              void* d_out, int out_size, void* d_ws, size_t ws_size,
                              hipStream_t stream) {
    const float* beta  = (const float*)d_in[0];
    const float* gamma = (const float*)d_in[1];
    const float* zi    = (const float*)d_in[2];
    const float* zj    = (const float*)d_in[3];
    const float* valC  = (const float*)d_in[4];
    const int*   si    = (const int*)d_in[5];
    const int*   sj    = (const int*)d_in[6];
    const int*   li    = (const int*)d_in[7];
    const int*   lj    = (const int*)d_in[8];
    float* out = (float*)d_out;
    const int nnz = in_sizes[4];

    // Workspace layout (floats): a_s, b_s, beta_s, gamma_s, na, nb
    float* ws      = (float*)d_ws;
    float* a_s     = ws;                    // SI_N*DIM
    float* b_s     = a_s + SI_N * DIM;      // SJ_N*DIM
    float* beta_s  = b_s + SJ_N * DIM;      // SI_N
    float* gamma_s = beta_s + SI_N;         // SJ_N
    float* na      = gamma_s + SJ_N;        // SI_N
    float* nb      = na + SI_N;             // SJ_N

    lsm_init_kernel<<<1, 1, 0, stream>>>(out);

    lsm_gather_kernel<<<(SI_N + 255) / 256, 256, 0, stream>>>(
        beta, gamma, zi, zj, si, sj, a_s, b_s, beta_s, gamma_s, na, nb);

    lsm_dense_kernel<<<dim3(I_TILES, J_BLOCKS), 32, 0, stream>>>(
        a_s, b_s, beta_s, gamma_s, na, nb, out);

    lsm_links_kernel<<<512, 256, 0, stream>>>(
        beta, gamma, zi, zj, valC, li, lj, out, nnz);
}